// IdAttn2d_55121610277068
// MI455X (gfx1250) — compile-verified
//
#include <hip/hip_runtime.h>
#include <hip/hip_bf16.h>

typedef __attribute__((ext_vector_type(2))) float v2f;
typedef __attribute__((ext_vector_type(8))) float v8f;

#define F_SIZE 256
#define LDSP 260  // padded LDS row stride (floats) to dodge bank conflicts

// ---------------------------------------------------------------------------
// K1: attention weights (GEMV + group-of-4 softmax) + pyramid id gather.
// One wave per active row; lane j owns output j = k*4 + p  (k=head, p=scale).
// ---------------------------------------------------------------------------
__global__ __launch_bounds__(256) void k_attn_ids(
    const float* __restrict__ act, const int* __restrict__ bids,
    const int* __restrict__ mids, const int* __restrict__ xy,
    const int* __restrict__ m0, const int* __restrict__ m1,
    const int* __restrict__ m2, const int* __restrict__ m3,
    const float* __restrict__ Wa, const float* __restrict__ ba,
    float* __restrict__ fw_ws, int* __restrict__ id_ws, int n_act)
{
    const int wave = threadIdx.x >> 5;
    const int lane = threadIdx.x & 31;
    const int row  = blockIdx.x * 8 + wave;
    if (row >= n_act) return;

    // --- aw[j] = act[row] . Wa[j] + ba[j] ---------------------------------
    const float4* __restrict__ arow = (const float4*)(act + (size_t)row * F_SIZE);
    const float4* __restrict__ wrow = (const float4*)(Wa + (size_t)lane * F_SIZE);
    float acc = 0.f;
#pragma unroll 8
    for (int i = 0; i < F_SIZE / 4; ++i) {
        float4 a = arow[i];
        float4 w = wrow[i];
        acc += a.x * w.x + a.y * w.y + a.z * w.z + a.w * w.w;
    }
    acc += ba[lane];

    // --- softmax over p (4 consecutive lanes share one head k) ------------
    float mx = acc;
    mx = fmaxf(mx, __shfl_xor(mx, 1, 32));
    mx = fmaxf(mx, __shfl_xor(mx, 2, 32));
    float e = __expf(acc - mx);
    float s = e;
    s += __shfl_xor(s, 1, 32);
    s += __shfl_xor(s, 2, 32);
    const float fw = e / s;

    // --- feat id for (p, k) ----------------------------------------------
    const int p = lane & 3;       // scale index 0..3
    const int k = lane >> 2;      // neighbor index 0..7
    const int offx[8] = {-1, 0, 1, -1, 1, -1, 0, 1};
    const int offy[8] = {-1, -1, -1, 0, 0, 1, 1, 1};
    const int scale = p + 1;
    int x = xy[row * 2 + 0] + scale * offx[k];
    int y = xy[row * 2 + 1] + scale * offy[k];
    const int m = mids[row];
    const int dim = 128 >> m;     // 128, 64, 32, 16
    x = min(max(x, 0), dim - 1);
    y = min(max(y, 0), dim - 1);
    const int* __restrict__ map = (m == 0) ? m0 : (m == 1) ? m1 : (m == 2) ? m2 : m3;
    const int b = bids[row];
    const int id = map[((size_t)b * dim + y) * dim + x];

    // store in gather order g = p*8 + k
    const size_t o = (size_t)row * 32 + p * 8 + k;
    fw_ws[o] = fw;
    id_ws[o] = id;
}

// ---------------------------------------------------------------------------
// K2: weighted gather-reduce.  One 256-thread block per row; thread t owns
// feature column t.  Rows are 1KB contiguous -> coalesced; the whole feature
// table (~123MB) is L2-resident on MI455X (192MB L2).
// ---------------------------------------------------------------------------
__global__ __launch_bounds__(256) void k_gather(
    const float* __restrict__ act, const float* __restrict__ pas,
    const float* __restrict__ fw_ws, const int* __restrict__ id_ws,
    float* __restrict__ weighted, int n_act)
{
    __shared__ float s_fw[32];
    __shared__ const float* s_ptr[32];
    const int row = blockIdx.x;
    const int t = threadIdx.x;
    if (t < 32) {
        const int id = id_ws[(size_t)row * 32 + t];
        s_fw[t] = fw_ws[(size_t)row * 32 + t];
        s_ptr[t] = (id < n_act) ? (act + (size_t)id * F_SIZE)
                                : (pas + (size_t)(id - n_act) * F_SIZE);
    }
    __syncthreads();
    float acc = 0.f;
#pragma unroll
    for (int g = 0; g < 32; ++g)
        acc += s_fw[g] * s_ptr[g][t];
    weighted[(size_t)row * F_SIZE + t] = acc;
}

// ---------------------------------------------------------------------------
// K3: fused  out = (weighted @ Wv^T + bv) @ Wo^T + bo  using fp32 WMMA
// (v_wmma_f32_16x16x4_f32).  Block = 256 threads = 8 waves, M-tile = 16 rows,
// each wave owns two 16-wide N-tiles (8*2*16 = 256 = full N).
// A-frag layout (ISA 16x4 f32): lane L -> M = L%16, holds K = k+2*(L/16)+{0,1}.
// B-frag mirrors it: lane L -> N = L%16 of W row, same K pair (W rows are
// contiguous in K so these are float2 loads).
// C/D: VGPR r of lane L -> (M = r + 8*(L/16), N = L%16).
// ---------------------------------------------------------------------------
__global__ __launch_bounds__(256) void k_dual_gemm(
    const float* __restrict__ weighted,
    const float* __restrict__ Wv, const float* __restrict__ bv,
    const float* __restrict__ Wo, const float* __restrict__ bo,
    float* __restrict__ out)
{
    __shared__ float lds_a[16 * LDSP];   // staged `weighted` M-tile
    __shared__ float lds_v[16 * LDSP];   // intermediate `val` M-tile

    const int t = threadIdx.x;
    const int wave = t >> 5;
    const int lane = t & 31;
    const int m0 = blockIdx.x * 16;

    // stage weighted[m0:m0+16, :] into LDS (each thread: one 64B chunk)
    {
        const int r = t >> 4;              // 0..15
        const int c0 = (t & 15) * 16;      // 0..240
        const float4* __restrict__ src =
            (const float4*)(weighted + (size_t)(m0 + r) * F_SIZE + c0);
        float4* dst = (float4*)(lds_a + r * LDSP + c0);
        dst[0] = src[0]; dst[1] = src[1]; dst[2] = src[2]; dst[3] = src[3];
    }
    __syncthreads();

    const int mrow = lane & 15;  // M for A-frag, N for B/C/D
    const int hi = lane >> 4;    // selects K pair within the 16x4 slice

    // ---- GEMM 1: val = lds_a @ Wv^T + bv  -> lds_v -----------------------
#pragma unroll
    for (int tn = 0; tn < 2; ++tn) {
        const int nn = (wave * 2 + tn) * 16 + mrow;
        const float* __restrict__ ap = lds_a + mrow * LDSP + 2 * hi;
        const float* __restrict__ bp = Wv + (size_t)nn * F_SIZE + 2 * hi;
        v8f c = {};
#pragma unroll 4
        for (int k = 0; k < F_SIZE; k += 4) {
            v2f a, b;
            a.x = ap[k]; a.y = ap[k + 1];
            b.x = bp[k]; b.y = bp[k + 1];
            c = __builtin_amdgcn_wmma_f32_16x16x4_f32(
                    false, a, false, b, (short)0, c, false, false);
        }
        const float bias = bv[nn];
#pragma unroll
        for (int r = 0; r < 8; ++r)
            lds_v[(r + 8 * hi) * LDSP + nn] = c[r] + bias;
    }
    __syncthreads();

    // ---- GEMM 2: out = lds_v @ Wo^T + bo ---------------------------------
#pragma unroll
    for (int tn = 0; tn < 2; ++tn) {
        const int nn = (wave * 2 + tn) * 16 + mrow;
        const float* __restrict__ ap = lds_v + mrow * LDSP + 2 * hi;
        const float* __restrict__ bp = Wo + (size_t)nn * F_SIZE + 2 * hi;
        v8f c = {};
#pragma unroll 4
        for (int k = 0; k < F_SIZE; k += 4) {
            v2f a, b;
            a.x = ap[k]; a.y = ap[k + 1];
            b.x = bp[k]; b.y = bp[k + 1];
            c = __builtin_amdgcn_wmma_f32_16x16x4_f32(
                    false, a, false, b, (short)0, c, false, false);
        }
        const float bias = bo[nn];
#pragma unroll
        for (int r = 0; r < 8; ++r)
            out[(size_t)(m0 + r + 8 * hi) * F_SIZE + nn] = c[r] + bias;
    }
}

// ---------------------------------------------------------------------------
extern "C" void kernel_launch(void* const* d_in, const int* in_sizes, int n_in,
                              void* d_out, int out_size, void* d_ws, size_t ws_size,
                              hipStream_t stream) {
    const float* act  = (const float*)d_in[0];
    const float* pas  = (const float*)d_in[1];
    const int*   bids = (const int*)d_in[2];
    const int*   mids = (const int*)d_in[3];
    const int*   xy   = (const int*)d_in[4];
    const int*   m0   = (const int*)d_in[5];
    const int*   m1   = (const int*)d_in[6];
    const int*   m2   = (const int*)d_in[7];
    const int*   m3   = (const int*)d_in[8];
    const float* Wa   = (const float*)d_in[9];
    const float* ba   = (const float*)d_in[10];
    const float* Wv   = (const float*)d_in[11];
    const float* bv   = (const float*)d_in[12];
    const float* Wo   = (const float*)d_in[13];
    const float* bo   = (const float*)d_in[14];
    float* out = (float*)d_out;

    const int n_act = in_sizes[0] / F_SIZE;   // 32768

    // workspace layout
    float* fw_ws    = (float*)d_ws;                       // n_act*32 f32
    int*   id_ws    = (int*)(fw_ws + (size_t)n_act * 32); // n_act*32 i32
    float* weighted = (float*)(id_ws + (size_t)n_act * 32); // n_act*256 f32

    k_attn_ids<<<(n_act + 7) / 8, 256, 0, stream>>>(
        act, bids, mids, xy, m0, m1, m2, m3, Wa, ba, fw_ws, id_ws, n_act);

    k_gather<<<n_act, 256, 0, stream>>>(act, pas, fw_ws, id_ws, weighted, n_act);

    k_dual_gemm<<<n_act / 16, 256, 0, stream>>>(weighted, Wv, bv, Wo, bo, out);
}